// ENMFLoss_63857573757113
// MI455X (gfx1250) — compile-verified
//
#include <hip/hip_runtime.h>
#include <hip/hip_bf16.h>

#define USER_NUM 200000
#define ITEM_NUM 100000
#define DIM 128
#define BSZ 4096
#define LLEN 200
#define NEG_W 0.1f

typedef __attribute__((ext_vector_type(2))) float v2f;
typedef __attribute__((ext_vector_type(8))) float v8f;

__device__ __forceinline__ v8f wmma_f32_4(v2f a, v2f b, v8f c) {
  // V_WMMA_F32_16X16X4_F32 : D(16x16,f32) = A(16x4,f32) * B(4x16,f32) + C
  return __builtin_amdgcn_wmma_f32_16x16x4_f32(
      /*neg_a=*/false, a, /*neg_b=*/false, b,
      /*c_mod=*/(short)0, c, /*reuse_a=*/false, /*reuse_b=*/false);
}

// ---------------------------------------------------------------------------
// Kernel 1: uh[b,d] = user_emb[uids[b],d] * h[d]; zero the two 128x128 accums.
// ---------------------------------------------------------------------------
__global__ void prep_kernel(const float* __restrict__ user_emb,
                            const float* __restrict__ h,
                            const int* __restrict__ uids,
                            float* __restrict__ uh,
                            float* __restrict__ P1,
                            float* __restrict__ P2) {
  int idx = blockIdx.x * blockDim.x + threadIdx.x;   // 0 .. BSZ*DIM-1
  int b = idx >> 7;
  int d = idx & (DIM - 1);
  uh[idx] = user_emb[(size_t)uids[b] * DIM + d] * h[d];
  if (idx < DIM * DIM) { P1[idx] = 0.0f; P2[idx] = 0.0f; }
}

// ---------------------------------------------------------------------------
// Kernel 2: K-split SYRK  out(128x128) += E[chunk]^T * E[chunk]  via f32 WMMA.
// Each workgroup: 1024 rows of K in 16 LDS-staged sub-chunks of 64.
// 8 waves; wave w owns the 128x16 column strip j=w (8 WMMA tiles).
// LDS tile stored d-major with stride 66 (bank-conflict-free b64 reads).
// ---------------------------------------------------------------------------
#define KSUB 64
#define KBLK 1024
#define LDS_STRIDE 66

__global__ void syrk_wmma_kernel(const float* __restrict__ E, int K,
                                 float* __restrict__ out) {
  __shared__ float lds[DIM * LDS_STRIDE];   // 33792 B

  const int tid  = threadIdx.x;
  const int lane = tid & 31;
  const int w    = tid >> 5;          // wave id 0..7 -> column strip
  const int lm   = lane & 15;         // row/col within tile
  const int khalf = (lane >> 4) << 1; // lanes 16..31 handle K+2,K+3

  v8f acc[8];
#pragma unroll
  for (int i = 0; i < 8; ++i) acc[i] = (v8f){0.f,0.f,0.f,0.f,0.f,0.f,0.f,0.f};

  for (int s = 0; s < KBLK / KSUB; ++s) {
    const int k0 = blockIdx.x * KBLK + s * KSUB;
    __syncthreads();   // protect previous sub-chunk's readers
    // Stage E[k0:k0+64, 0:128] transposed to d-major: lds[d*66 + kk]
    for (int idx = tid; idx < KSUB * DIM; idx += 256) {
      int d  = idx & (DIM - 1);
      int kk = idx >> 7;
      int k  = k0 + kk;
      lds[d * LDS_STRIDE + kk] = (k < K) ? E[(size_t)k * DIM + d] : 0.0f;
    }
    __syncthreads();

    for (int kk = 0; kk < KSUB; kk += 4) {
      const int ko = kk + khalf;  // even -> 8B-aligned ds_load_b64
      v2f bfrag = *(const v2f*)&lds[(16 * w + lm) * LDS_STRIDE + ko];
#pragma unroll
      for (int i = 0; i < 8; ++i) {
        v2f afrag = *(const v2f*)&lds[(16 * i + lm) * LDS_STRIDE + ko];
        acc[i] = wmma_f32_4(afrag, bfrag, acc[i]);
      }
    }
  }

  // C/D layout: VGPR r holds row M=r (lanes 0-15) / M=8+r (lanes 16-31)
  const int rbase = (lane >> 4) << 3;
  const int col   = 16 * w + lm;
#pragma unroll
  for (int i = 0; i < 8; ++i) {
#pragma unroll
    for (int r = 0; r < 8; ++r) {
      int row = 16 * i + rbase + r;
      atomicAdd(&out[row * DIM + col], acc[i][r]);
    }
  }
}

// ---------------------------------------------------------------------------
// Kernel 3: Sc = NEG_W * sum(P1 .* P2h) / B   (single workgroup)
// ---------------------------------------------------------------------------
__global__ void reduce_S_kernel(const float* __restrict__ P1,
                                const float* __restrict__ P2,
                                float* __restrict__ Sc) {
  __shared__ float red[256];
  float s = 0.0f;
  for (int i = threadIdx.x; i < DIM * DIM; i += 256) s += P1[i] * P2[i];
  red[threadIdx.x] = s;
  __syncthreads();
  for (int off = 128; off > 0; off >>= 1) {
    if (threadIdx.x < off) red[threadIdx.x] += red[threadIdx.x + off];
    __syncthreads();
  }
  if (threadIdx.x == 0) Sc[0] = NEG_W * red[0] / (float)BSZ;
}

// ---------------------------------------------------------------------------
// Kernel 4: gather-dot.  One wave per batch row b.
//   hpq = <item_emb[iid], uh[b]>;  loss += 0.9*hpq^2 - 2*hpq  (skip pads)
//   out[b] = loss + Sc
// Rows are 512B coalesced loads; table is L2-resident (51MB < 192MB L2).
// ---------------------------------------------------------------------------
__global__ void pos_loss_kernel(const float* __restrict__ item_emb,
                                const int* __restrict__ pos_iids,
                                const float* __restrict__ uh,
                                const float* __restrict__ Sc,
                                float* __restrict__ out) {
  const int wid  = (blockIdx.x * blockDim.x + threadIdx.x) >> 5;
  const int lane = threadIdx.x & 31;
  const int b = wid;

  const float4 u = ((const float4*)(uh + (size_t)b * DIM))[lane];
  const int* row = pos_iids + (size_t)b * LLEN;

  float loss = 0.0f;
  for (int lb = 0; lb < LLEN; lb += 32) {
    const int li = lb + lane;
    int myiid = (li < LLEN) ? row[li] : ITEM_NUM;
#pragma unroll 4
    for (int t = 0; t < 32; ++t) {
      if (lb + t >= LLEN) break;
      int iid = __shfl(myiid, t, 32);
      if (iid == ITEM_NUM) continue;           // masked / padding row
      const float4 r = ((const float4*)(item_emb + (size_t)iid * DIM))[lane];
      float p = fmaf(r.x, u.x, fmaf(r.y, u.y, fmaf(r.z, u.z, r.w * u.w)));
#pragma unroll
      for (int off = 16; off > 0; off >>= 1) p += __shfl_xor(p, off, 32);
      loss += (1.0f - NEG_W) * p * p - 2.0f * p;
    }
  }
  if (lane == 0) out[b] = loss + Sc[0];
}

// ---------------------------------------------------------------------------
extern "C" void kernel_launch(void* const* d_in, const int* in_sizes, int n_in,
                              void* d_out, int out_size, void* d_ws, size_t ws_size,
                              hipStream_t stream) {
  const float* user_emb = (const float*)d_in[0];   // [USER_NUM, DIM]
  const float* item_emb = (const float*)d_in[1];   // [ITEM_NUM+1, DIM]
  const float* h        = (const float*)d_in[2];   // [DIM, 1]
  const int*   uids     = (const int*)d_in[3];     // [B]
  const int*   pos_iids = (const int*)d_in[4];     // [B, L]
  float* out = (float*)d_out;                      // [B]

  float* ws = (float*)d_ws;
  float* P1  = ws;                                 // [128*128]
  float* P2h = ws + DIM * DIM;                     // [128*128]
  float* uh  = ws + 2 * DIM * DIM;                 // [B*128]
  float* Sc  = uh + (size_t)BSZ * DIM;             // [1]

  // 1) uh = gather(user_emb)[b]*h ; zero accumulators
  prep_kernel<<<(BSZ * DIM) / 256, 256, 0, stream>>>(user_emb, h, uids, uh, P1, P2h);

  // 2) P1 += item_emb^T item_emb  (K = ITEM_NUM+1)
  const int K1 = ITEM_NUM + 1;
  syrk_wmma_kernel<<<(K1 + KBLK - 1) / KBLK, 256, 0, stream>>>(item_emb, K1, P1);

  // 3) P2h += uh^T uh  (K = B)  — h folded in, replaces part_2 .* part_3
  syrk_wmma_kernel<<<(BSZ + KBLK - 1) / KBLK, 256, 0, stream>>>(uh, BSZ, P2h);

  // 4) Sc = NEG_W * sum(P1 .* P2h)/B
  reduce_S_kernel<<<1, 256, 0, stream>>>(P1, P2h, Sc);

  // 5) per-b positive-sample loss + broadcast Sc
  pos_loss_kernel<<<BSZ / 8, 256, 0, stream>>>(item_emb, pos_iids, uh, Sc, out);
}